// FKLoss_66151086293043
// MI455X (gfx1250) — compile-verified
//
#include <hip/hip_runtime.h>
#include <hip/hip_bf16.h>
#include <math.h>

// ---------------------------------------------------------------------------
// FK loss for 6-DOF arm (UR5-style), B = 524288.
// Memory floor: 25 MB / 23.3 TB/s ~ 1.1us. Compute: ~500 f32 flops + 12
// sin/cos + 3 atan2 per element -> VALU/trans bound, far below any WMMA
// regime for the 4x4 chains (25% diag-block utilization + cross-lane
// re-layout per step would cost more than the 36-FMA affine product).
// WMMA f32 16x16x4 is used where it is exact and cheap: the 32-lane wave
// reduction (A = partials, B = ones -> row sums), one matrix op instead of
// a 5-step shuffle tree. Two-kernel deterministic reduction (no atomics).
// ---------------------------------------------------------------------------

typedef float v2f __attribute__((ext_vector_type(2)));
typedef float v8f __attribute__((ext_vector_type(8)));

// Wave32 sum via V_WMMA_F32_16X16X4_F32.
// A layout (16x4 f32): v0 lanes0-15 = (M=lane,K=0), lanes16-31 = (M,K=2);
// v1 = K=1/K=3.  Set a0=p, a1=0  ->  A[m][0]=p_m, A[m][2]=p_{m+16}.
// B = ones -> C[m][n] = p_m + p_{m+16}.
// C layout: vgpr r lanes0-15 = row r, lanes16-31 = row r+8. Summing the 8
// C vgprs gives lanes0-15: sum(p_0..7,p_16..23), lanes16-31: the rest;
// one xor-16 shuffle completes the total (broadcast to all lanes).
__device__ __forceinline__ float wave_sum_wmma(float p) {
    v2f a; a[0] = p;    a[1] = 0.0f;
    v2f b; b[0] = 1.0f; b[1] = 1.0f;
    v8f c = {};
    c = __builtin_amdgcn_wmma_f32_16x16x4_f32(
            /*neg_a=*/false, a, /*neg_b=*/false, b,
            /*c_mod=*/(short)0, c, /*reuse_a=*/false, /*reuse_b=*/false);
    float t = ((c[0] + c[1]) + (c[2] + c[3])) + ((c[4] + c[5]) + (c[6] + c[7]));
    t += __shfl_xor(t, 16, 32);
    return t;
}

// M (3x4 affine, implied bottom row [0,0,0,1]) <- M * T (3x4 affine)
__device__ __forceinline__ void mul34(float* M, const float* T) {
#pragma unroll
    for (int i = 0; i < 3; ++i) {
        float a = M[4*i+0], b = M[4*i+1], c = M[4*i+2], d = M[4*i+3];
        M[4*i+0] = a*T[0] + b*T[4] + c*T[8];
        M[4*i+1] = a*T[1] + b*T[5] + c*T[9];
        M[4*i+2] = a*T[2] + b*T[6] + c*T[10];
        M[4*i+3] = a*T[3] + b*T[7] + c*T[11] + d;
    }
}

__device__ __forceinline__ float fk_loss6(const float* ja, const float* g) {
    float s[6], c[6];
#pragma unroll
    for (int j = 0; j < 6; ++j) {
        s[j] = __sinf(ja[j]);   // v_sin_f32 (args ~N(0,1): native range OK)
        c[j] = __cosf(ja[j]);
    }

    // M = BASE * T0   (BASE = diag(-1,-1,1,1) -> negate rows 0,1 of T0)
    float M[12] = { -c[0], 0.f, -s[0], 0.f,
                    -s[0], 0.f,  c[0], 0.f,
                     0.f,  1.f,  0.f,  0.089159f };

    { const float a = -0.425f;   // T1
      float T[12] = { c[1], -s[1], 0.f, a*c[1],
                      s[1],  c[1], 0.f, a*s[1],
                      0.f,   0.f,  1.f, 0.f };
      mul34(M, T); }
    { const float a = -0.39225f; // T2
      float T[12] = { c[2], -s[2], 0.f, a*c[2],
                      s[2],  c[2], 0.f, a*s[2],
                      0.f,   0.f,  1.f, 0.f };
      mul34(M, T); }
    { float T[12] = { c[3], 0.f,  s[3], 0.f,   // T3
                      s[3], 0.f, -c[3], 0.f,
                      0.f,  1.f,  0.f,  0.10915f };
      mul34(M, T); }
    { float T[12] = { c[4], 0.f, -s[4], 0.f,   // T4
                      s[4], 0.f,  c[4], 0.f,
                      0.f, -1.f,  0.f,  0.09465f };
      mul34(M, T); }
    { float T[12] = { c[5], -s[5], 0.f, 0.f,   // T5
                      s[5],  c[5], 0.f, 0.f,
                      0.f,   0.f,  1.f, 0.0823f };
      mul34(M, T); }

    // Fold LAST: col0 = Mcol2, col1 = -Mcol0, col2 = -Mcol1, col3 = Mcol3
    float R00 =  M[2],  R10 =  M[6],  R20 =  M[10];
    float R11 = -M[4],  R12 = -M[5];
    float R21 = -M[8],  R22 = -M[9];
    float tx = M[3], ty = M[7], tz = M[11];

    float sy = sqrtf(R00*R00 + R10*R10);
    bool sing = sy < 1e-6f;
    float x = sing ? atan2f(-R12, R11) : atan2f(R21, R22);
    float y = atan2f(-R20, sy);
    float z = sing ? 0.f : atan2f(R10, R00);

    return fabsf(x - g[0]) + fabsf(y - g[1]) + fabsf(z - g[2]) +
           fabsf(tx - g[3]) + fabsf(ty - g[4]) + fabsf(tz - g[5]);
}

__global__ void __launch_bounds__(256)
fk_partial_kernel(const float* __restrict__ ja, const float* __restrict__ gt,
                  float* __restrict__ partial, int B) {
    const int stride = gridDim.x * blockDim.x;
    float p = 0.f;
    for (int e = blockIdx.x * blockDim.x + threadIdx.x; e < B; e += stride) {
        const float* jp = ja + 6 * (size_t)e;
        const float* gp = gt + 6 * (size_t)e;
        __builtin_prefetch(jp + 6 * (size_t)stride, 0, 1);
        __builtin_prefetch(gp + 6 * (size_t)stride, 0, 1);
        float j6[6], g6[6];
        // 24B element stride is 8-aligned -> float2 loads
        *(float2*)&j6[0] = ((const float2*)jp)[0];
        *(float2*)&j6[2] = ((const float2*)jp)[1];
        *(float2*)&j6[4] = ((const float2*)jp)[2];
        *(float2*)&g6[0] = ((const float2*)gp)[0];
        *(float2*)&g6[2] = ((const float2*)gp)[1];
        *(float2*)&g6[4] = ((const float2*)gp)[2];
        p += fk_loss6(j6, g6);
    }

    // wave reduce via WMMA (EXEC all-ones here: loop has reconverged)
    float w = wave_sum_wmma(p);

    __shared__ float sp[8];
    const int lane = threadIdx.x & 31;
    const int wid  = threadIdx.x >> 5;
    if (lane == 0) sp[wid] = w;
    __syncthreads();
    if (threadIdx.x == 0) {
        float sum = 0.f;
#pragma unroll
        for (int i = 0; i < 8; ++i) sum += sp[i];
        partial[blockIdx.x] = sum;
    }
}

__global__ void __launch_bounds__(256)
fk_final_kernel(const float* __restrict__ partial, int n,
                float* __restrict__ out, float inv_count) {
    float p = 0.f;
    for (int i = threadIdx.x; i < n; i += blockDim.x) p += partial[i];

    float w = wave_sum_wmma(p);

    __shared__ float sp[8];
    const int lane = threadIdx.x & 31;
    const int wid  = threadIdx.x >> 5;
    if (lane == 0) sp[wid] = w;
    __syncthreads();
    if (threadIdx.x == 0) {
        float sum = 0.f;
#pragma unroll
        for (int i = 0; i < 8; ++i) sum += sp[i];
        out[0] = sum * inv_count;
    }
}

extern "C" void kernel_launch(void* const* d_in, const int* in_sizes, int n_in,
                              void* d_out, int out_size, void* d_ws, size_t ws_size,
                              hipStream_t stream) {
    const float* ja = (const float*)d_in[0];
    const float* gt = (const float*)d_in[1];
    const int B = in_sizes[0] / 6;

    float* partial = (float*)d_ws;

    const int block = 256;                      // 8 wave32s per workgroup
    int grid = (B + block * 4 - 1) / (block * 4); // ~4 elements per thread
    if (grid < 1) grid = 1;
    if (grid > 1024) grid = 1024;               // ws partials stay tiny

    fk_partial_kernel<<<grid, block, 0, stream>>>(ja, gt, partial, B);
    fk_final_kernel<<<1, block, 0, stream>>>(partial, grid, (float*)d_out,
                                             1.0f / (6.0f * (float)B));
}